// BilevelEncoder_59931973649042
// MI455X (gfx1250) — compile-verified
//
#include <hip/hip_runtime.h>
#include <math.h>

typedef float v2f __attribute__((ext_vector_type(2)));
typedef float v8f __attribute__((ext_vector_type(8)));

#define NRES 256
#define BSZ  4
#define PGC  64
#define LIG  32
#define HCD  128
#define ECD  64
#define NH   4
#define DH   32
#define NCA  14
#define KNN_ 8
#define NEDGE (NRES*KNN_+NRES)      /* 2304 */
#define NPAIR (NEDGE*NCA*NCA)       /* 451584 */
#define E1C  64
#define NP1  (E1C*NCA*LIG)          /* 28672 */
#define LOG2F_ 0.6931471805599453f

__constant__ int c_counts[21] = {14,5,11,8,8,6,9,9,4,10,8,8,9,8,11,7,6,7,14,12,7};

__device__ __forceinline__ float amask_of(int s, int p){
  return (p < c_counts[s]) ? 1.f : 0.f;
}
__device__ __forceinline__ float sspf(float x){
  float sp = (x > 15.f) ? x : log1pf(__expf(x));
  return sp - LOG2F_;
}
__device__ __forceinline__ float rbff(float d, int k){
  const float step = 10.0f/(float)(ECD-1);
  const float coeff = -0.5f/(step*step);
  float t = d - (float)k*step;
  return __expf(coeff*t*t);
}
__device__ __forceinline__ void edge_rc(int e, const int* nbr, int& r, int& c){
  if(e < NRES*KNN_){ r = e/KNN_; c = nbr[e]; } else { r = e - NRES*KNN_; c = r; }
}
__device__ __forceinline__ v8f wmma4(v2f a, v2f b, v8f c){
  // V_WMMA_F32_16X16X4_F32 : D = A(16x4) x B(4x16) + C(16x16), all fp32
  return __builtin_amdgcn_wmma_f32_16x16x4_f32(false, a, false, b, (short)0, c, false, false);
}

// ---------------------------------------------------------------- KNN (per batch group of 64)
__global__ void knn_kernel(const float* __restrict__ resX, int* __restrict__ nbr){
  int i = blockIdx.x*blockDim.x + threadIdx.x;
  if(i >= NRES) return;
  int base = (i/PGC)*PGC;
  float cx = resX[(i*NCA+1)*3+0];
  float cy = resX[(i*NCA+1)*3+1];
  float cz = resX[(i*NCA+1)*3+2];
  float bd[KNN_]; int bj[KNN_];
  for(int k=0;k<KNN_;k++){ bd[k]=3.0e38f; bj[k]=0; }
  for(int j=base;j<base+PGC;j++){
    if(j==i) continue;
    float dx=resX[(j*NCA+1)*3+0]-cx;
    float dy=resX[(j*NCA+1)*3+1]-cy;
    float dz=resX[(j*NCA+1)*3+2]-cz;
    float d2=dx*dx+dy*dy+dz*dz;
    if(d2 < bd[KNN_-1]){
      int k=KNN_-1;
      while(k>0 && bd[k-1]>d2){ bd[k]=bd[k-1]; bj[k]=bj[k-1]; k--; }
      bd[k]=d2; bj[k]=j;
    }
  }
  for(int k=0;k<KNN_;k++) nbr[i*KNN_+k]=bj[k];
}

// ---------------------------------------------------------------- LayerNorm (one row / block of 128)
__global__ void ln_kernel(const float* __restrict__ x, const float* __restrict__ g,
                          const float* __restrict__ b, float* __restrict__ out){
  int row = blockIdx.x, t = threadIdx.x;
  float v = x[row*HCD + t];
  __shared__ float red[HCD];
  red[t]=v; __syncthreads();
  for(int s=HCD/2;s>0;s>>=1){ if(t<s) red[t]+=red[t+s]; __syncthreads(); }
  float mu = red[0]*(1.f/HCD);
  __syncthreads();
  float dv = v-mu;
  red[t]=dv*dv; __syncthreads();
  for(int s=HCD/2;s>0;s>>=1){ if(t<s) red[t]+=red[t+s]; __syncthreads(); }
  float var = red[0]*(1.f/HCD);
  out[row*HCD+t] = dv*rsqrtf(var+1e-5f)*g[t]+b[t];
}

// ---------------------------------------------------------------- generic NT GEMM: out[M,128] = A[M,128] @ W[:,koff:koff+128].T
// mode 0: store; mode 1: store addsrc+acc; mode 2: scatter-add (res_H3 rows via row1 mapping)
__global__ void gemm_nt(const float* __restrict__ A, const float* __restrict__ W,
                        float* __restrict__ out, const float* __restrict__ addsrc,
                        int M, int ldw, int koff, int mode){
  int wave = threadIdx.x >> 5, lane = threadIdx.x & 31;
  int tile = blockIdx.x*4 + wave;
  int tiles = (M>>4)*8;
  if(tile >= tiles) return;
  int tm = tile>>3, tn = tile&7;
  int half = lane>>4, lm = lane&15;
  const float* Ar = A + (size_t)(tm*16+lm)*HCD;
  const float* Wr = W + (size_t)(tn*16+lm)*ldw + koff;
  v8f acc = {0,0,0,0,0,0,0,0};
  #pragma unroll
  for(int kk=0;kk<HCD;kk+=4){
    int k0 = kk + half*2;
    v2f a = *(const v2f*)(Ar + k0);
    v2f b = *(const v2f*)(Wr + k0);
    acc = wmma4(a,b,acc);
  }
  #pragma unroll
  for(int v=0;v<8;v++){
    int m = tm*16 + v + half*8;
    int n = tn*16 + lm;
    float val = acc[v];
    if(mode==0) out[(size_t)m*HCD+n]=val;
    else if(mode==1) out[(size_t)m*HCD+n]=addsrc[(size_t)m*HCD+n]+val;
    else {
      int j=m/NCA, p=m-j*NCA;
      int orow = ((j>>4)*PGC + (j&15))*NCA + p;  // row1[j]*14+p
      out[(size_t)orow*HCD+n] += val;
    }
  }
}

// ---------------------------------------------------------------- fused RBF->MLP pair kernel (one wave = 16 pairs)
// mode 0: sig (res-res, out [E,H,14,14]); mode 1: fpre (adds Tq/Tk, out [E,196]);
// mode 2: sig1 (res-lig, out [e1,H,14,32])
__global__ void pair_mlp(int mode, const float* __restrict__ resX,
                         const float* __restrict__ lpos, const int* __restrict__ nbr,
                         const float* __restrict__ w1, int ldw1, int koff1,
                         const float* __restrict__ b1,
                         const float* __restrict__ w2, const float* __restrict__ b2,
                         const float* __restrict__ tq, const float* __restrict__ tk,
                         float* __restrict__ outbuf)
{
  int tilebase = blockIdx.x*16;
  int lane = threadIdx.x;
  int half = lane>>4, lm = lane&15;
  int pidxA = tilebase + lm;
  float dx,dy,dz;
  if(mode==2){
    int j = pidxA/(NCA*LIG), pq = pidxA%(NCA*LIG);
    int p = pq/LIG, q = pq - p*LIG;
    int ri = ((j>>4)*PGC + (j&15));
    int bb = j>>4;
    dx = resX[(ri*NCA+p)*3+0]-lpos[(bb*LIG+q)*3+0];
    dy = resX[(ri*NCA+p)*3+1]-lpos[(bb*LIG+q)*3+1];
    dz = resX[(ri*NCA+p)*3+2]-lpos[(bb*LIG+q)*3+2];
  } else {
    int e = pidxA/(NCA*NCA), pq = pidxA%(NCA*NCA);
    int p=pq/NCA, q=pq-p*NCA;
    int r,c; edge_rc(e,nbr,r,c);
    dx = resX[(r*NCA+p)*3+0]-resX[(c*NCA+q)*3+0];
    dy = resX[(r*NCA+p)*3+1]-resX[(c*NCA+q)*3+1];
    dz = resX[(r*NCA+p)*3+2]-resX[(c*NCA+q)*3+2];
  }
  float d = sqrtf(fmaxf(dx*dx+dy*dy+dz*dz, 1e-12f));  // _safe_norm

  const v8f z8 = {0,0,0,0,0,0,0,0};
  v8f acc[8];
  #pragma unroll
  for(int t=0;t<8;t++) acc[t]=z8;
  for(int kk=0;kk<ECD;kk+=4){
    int k0=kk+half*2;
    v2f a; a.x=rbff(d,k0); a.y=rbff(d,k0+1);
    #pragma unroll
    for(int t=0;t<8;t++){
      v2f b = *(const v2f*)(w1 + (size_t)(t*16+lm)*ldw1 + koff1 + k0);
      acc[t]=wmma4(a,b,acc[t]);
    }
  }
  __shared__ float hid[16*132];
  int rq[8], rk[8];
  if(mode==1){
    #pragma unroll
    for(int v=0;v<8;v++){
      int pr = tilebase + v + half*8;
      int e=pr/(NCA*NCA), pq=pr%(NCA*NCA);
      int p=pq/NCA, q=pq-p*NCA;
      int r,c; edge_rc(e,nbr,r,c);
      rq[v]=(r*NCA+p)*HCD; rk[v]=(c*NCA+q)*HCD;
    }
  }
  #pragma unroll
  for(int t=0;t<8;t++){
    int jc = t*16+lm;
    float bias = b1[jc];
    #pragma unroll
    for(int v=0;v<8;v++){
      float val = acc[t][v] + bias;
      if(mode==1) val += tq[rq[v]+jc] + tk[rk[v]+jc];
      val = fmaxf(val, 0.f);
      hid[(v+half*8)*132 + jc] = val;
    }
  }
  __syncthreads();
  if(mode==1){
    if(lane<16){
      float o = b2[0];
      for(int jj=0;jj<HCD;jj++) o += hid[lane*132+jj]*w2[jj];
      outbuf[tilebase+lane]=o;
    }
  } else {
    int m = lm;
    int h0 = half*2;
    float o0=b2[h0], o1=b2[h0+1];
    for(int jj=0;jj<HCD;jj++){
      float hv = hid[m*132+jj];
      o0 += hv*w2[h0*HCD+jj];
      o1 += hv*w2[(h0+1)*HCD+jj];
    }
    int pidx = tilebase+m;
    if(mode==0){
      int e=pidx/(NCA*NCA), pq=pidx%(NCA*NCA); int p=pq/NCA,q=pq-p*NCA;
      outbuf[(size_t)((e*NH+h0  )*NCA+p)*NCA+q]=o0;
      outbuf[(size_t)((e*NH+h0+1)*NCA+p)*NCA+q]=o1;
    } else {
      int j=pidx/(NCA*LIG), pq=pidx%(NCA*LIG); int p=pq/LIG,q=pq-p*LIG;
      outbuf[(size_t)((j*NH+h0  )*NCA+p)*LIG+q]=o0;
      outbuf[(size_t)((j*NH+h0+1)*NCA+p)*LIG+q]=o1;
    }
  }
}

// ---------------------------------------------------------------- per-edge residue attention (one wave / edge)
__global__ void edge_attn(const float* __restrict__ Qb, const float* __restrict__ Kb,
                          const float* __restrict__ Vb, const float* __restrict__ sigT,
                          const float* __restrict__ fpre, const float* __restrict__ resX,
                          const int* __restrict__ nbr, const int* __restrict__ resS,
                          float* __restrict__ rbuf, float* __restrict__ aVbuf,
                          float* __restrict__ fXbuf)
{
  int e = blockIdx.x;
  int lane = threadIdx.x, half=lane>>4, lm=lane&15;
  int r,c; edge_rc(e,nbr,r,c);
  __shared__ float amr[16], amc[16], lg[256], att[256], matt[256], fXa[48];
  if(lane<16){
    amr[lane] = (lane<NCA)? amask_of(resS[r],lane):0.f;
    amc[lane] = (lane<NCA)? amask_of(resS[c],lane):0.f;
  }
  for(int idx=lane; idx<256; idx+=32) matt[idx]=0.f;
  for(int idx=lane; idx<48; idx+=32) fXa[idx]=0.f;
  __syncthreads();
  float sr=0.f,sc=0.f;
  for(int p=0;p<NCA;p++){ sr+=amr[p]; sc+=amc[p]; }
  const float inv = 0.17677669529663687f; // 1/sqrt(32)
  const v8f z8 = {0,0,0,0,0,0,0,0};
  for(int h=0; h<NH; h++){
    v8f acc = z8;
    const float* Qr = Qb + (size_t)r*NCA*HCD + h*DH;
    const float* Kr = Kb + (size_t)c*NCA*HCD + h*DH;
    #pragma unroll
    for(int kk=0;kk<DH;kk+=4){
      int k0=kk+half*2;
      v2f a, b;
      if(lm<NCA) a = *(const v2f*)(Qr + lm*HCD + k0); else {a.x=0.f;a.y=0.f;}
      if(lm<NCA) b = *(const v2f*)(Kr + lm*HCD + k0); else {b.x=0.f;b.y=0.f;}
      acc = wmma4(a,b,acc);
    }
    #pragma unroll
    for(int v=0;v<8;v++){
      int p=v+half*8, q=lm;
      float val = 0.f;
      if(p<NCA && q<NCA) val = acc[v]*inv + sigT[(size_t)((e*NH+h)*NCA+p)*NCA+q];
      lg[p*16+q]=val;
    }
    __syncthreads();
    // r[e,h] = masked mean of logits
    float part=0.f;
    for(int idx=lane; idx<NCA*NCA; idx+=32){ int p=idx/NCA,q=idx-p*NCA; part += amr[p]*lg[p*16+q]*amc[q]; }
    for(int s=16;s>0;s>>=1) part += __shfl_xor(part, s, 32);
    if(lane==0) rbuf[e*NH+h]=part/(sr*sc);
    // row softmax -> masked -> L1 renorm
    if(lane<16){
      int p=lane;
      if(p<NCA){
        float mx=-3.0e38f;
        for(int q=0;q<NCA;q++) mx = fmaxf(mx, lg[p*16+q]);
        float s=0.f, s2=0.f;
        for(int q=0;q<NCA;q++){ float ex=__expf(lg[p*16+q]-mx); s+=ex; s2+=ex*amc[q]; }
        float den = amr[p]*s2/s + 1e-7f;
        float scale = amr[p]/(s*den);
        for(int q=0;q<NCA;q++) att[p*16+q] = __expf(lg[p*16+q]-mx)*amc[q]*scale;
        att[p*16+14]=0.f; att[p*16+15]=0.f;
      } else {
        for(int q=0;q<16;q++) att[p*16+q]=0.f;
      }
    }
    __syncthreads();
    for(int idx=lane; idx<256; idx+=32) matt[idx]+=att[idx];
    // aV = att @ V_h  (M=16,K=16pad,N=32)
    #pragma unroll
    for(int t=0;t<2;t++){
      v8f ac2 = z8;
      #pragma unroll
      for(int kk=0;kk<16;kk+=4){
        int k0=kk+half*2;
        v2f a; a.x=att[lm*16+k0]; a.y=att[lm*16+k0+1];
        int n=t*16+lm;
        v2f b;
        b.x = (k0  <NCA)? Vb[(size_t)(c*NCA+k0  )*HCD + h*DH + n] : 0.f;
        b.y = (k0+1<NCA)? Vb[(size_t)(c*NCA+k0+1)*HCD + h*DH + n] : 0.f;
        ac2 = wmma4(a,b,ac2);
      }
      #pragma unroll
      for(int v=0;v<8;v++){
        int p=v+half*8;
        if(p<NCA) aVbuf[(size_t)(e*NCA+p)*HCD + h*DH + t*16+lm] = ac2[v];
      }
    }
    __syncthreads();
  }
  // fX[e,p,:] = sum_q fpre*meanatt*Xij_hat
  for(int idx=lane; idx<NCA*NCA; idx+=32){
    int p=idx/NCA, q=idx-p*NCA;
    float f = fpre[(size_t)e*NCA*NCA+idx]*(matt[p*16+q]*0.25f);
    float dx = resX[(r*NCA+p)*3+0]-resX[(c*NCA+q)*3+0];
    float dy = resX[(r*NCA+p)*3+1]-resX[(c*NCA+q)*3+1];
    float dz = resX[(r*NCA+p)*3+2]-resX[(c*NCA+q)*3+2];
    float nrm = sqrtf(fmaxf(dx*dx+dy*dy+dz*dz,1e-12f)) + 1e-5f;
    float s = f/nrm;
    atomicAdd(&fXa[p*3+0], s*dx);
    atomicAdd(&fXa[p*3+1], s*dy);
    atomicAdd(&fXa[p*3+2], s*dz);
  }
  __syncthreads();
  for(int idx=lane; idx<NCA*3; idx+=32) fXbuf[(size_t)e*NCA*3+idx]=fXa[idx];
}

// ---------------------------------------------------------------- beta = segment softmax of r over row (9 edges / residue)
__global__ void beta_kernel(const float* __restrict__ rbuf, float* __restrict__ beta){
  int i = blockIdx.x*blockDim.x + threadIdx.x;
  if(i>=NRES) return;
  int ed[9];
  for(int k=0;k<8;k++) ed[k]=i*8+k;
  ed[8]=NRES*KNN_+i;
  for(int h=0;h<NH;h++){
    float mx=-3e38f;
    for(int k=0;k<9;k++) mx=fmaxf(mx, rbuf[ed[k]*NH+h]);
    float s=0.f;
    for(int k=0;k<9;k++) s+=__expf(rbuf[ed[k]*NH+h]-mx);
    float invs=1.f/s;
    for(int k=0;k<9;k++) beta[ed[k]*NH+h]=__expf(rbuf[ed[k]*NH+h]-mx)*invs;
  }
}

// ---------------------------------------------------------------- updH (ssp) and res_X2 (clip+mask)
__global__ void updh_dx(const float* __restrict__ aVbuf, const float* __restrict__ beta,
                        const float* __restrict__ fXbuf, const float* __restrict__ resX,
                        float* __restrict__ sspUH, float* __restrict__ outX){
  int i=blockIdx.x, t=threadIdx.x;
  int ed[9];
  for(int k=0;k<8;k++) ed[k]=i*8+k;
  ed[8]=NRES*KNN_+i;
  int h=t>>5;
  float bw[9];
  for(int k=0;k<9;k++) bw[k]=beta[ed[k]*NH+h];
  for(int p=0;p<NCA;p++){
    float s=0.f;
    for(int k=0;k<9;k++) s += bw[k]*aVbuf[(size_t)(ed[k]*NCA+p)*HCD + t];
    sspUH[(size_t)(i*NCA+p)*HCD+t]=sspf(s);
  }
  if(t<NCA*3){
    float s=0.f;
    for(int k=0;k<9;k++){
      float mb=0.25f*(beta[ed[k]*4]+beta[ed[k]*4+1]+beta[ed[k]*4+2]+beta[ed[k]*4+3]);
      s += mb*fXbuf[(size_t)ed[k]*NCA*3+t];
    }
    s = fminf(fmaxf(s,-3.f),3.f);
    float base = resX[i*NCA*3+t];
    bool rm = (i%PGC)<16;
    outX[i*NCA*3+t] = rm ? base+s : base;
  }
}

// ---------------------------------------------------------------- per-edge residue<->ligand attention (one wave / edge)
__global__ void lig_attn(const float* __restrict__ Qb, const float* __restrict__ Klb,
                         const float* __restrict__ Vlb, const float* __restrict__ Vb,
                         const float* __restrict__ sig1T, const float* __restrict__ lmask,
                         const int* __restrict__ resS,
                         float* __restrict__ r1buf, float* __restrict__ sspAVr,
                         float* __restrict__ aVlg)
{
  int j=blockIdx.x, lane=threadIdx.x, half=lane>>4, lm=lane&15;
  int b=j>>4, ri=b*PGC+(j&15);
  __shared__ float amr[16], lmq[32], lg[16*32], arr[16*32], alT[32*16], rden[16];
  if(lane<16) amr[lane] = (lane<NCA)? amask_of(resS[ri],lane) : 0.f;
  lmq[lane] = lmask[b*LIG+lane];
  __syncthreads();
  float sr=0.f; for(int p=0;p<NCA;p++) sr+=amr[p];
  float sc=0.f; for(int q=0;q<LIG;q++) sc+=lmq[q];
  const float inv=0.17677669529663687f;
  const v8f z8 = {0,0,0,0,0,0,0,0};
  for(int h=0;h<NH;h++){
    // logits 16x32
    #pragma unroll
    for(int t=0;t<2;t++){
      v8f acc=z8;
      #pragma unroll
      for(int kk=0;kk<DH;kk+=4){
        int k0=kk+half*2;
        v2f a,bv;
        if(lm<NCA) a=*(const v2f*)(Qb+(size_t)(ri*NCA+lm)*HCD+h*DH+k0); else{a.x=0.f;a.y=0.f;}
        int n=t*16+lm;
        bv=*(const v2f*)(Klb+(size_t)(b*LIG+n)*HCD+h*DH+k0);
        acc=wmma4(a,bv,acc);
      }
      #pragma unroll
      for(int v=0;v<8;v++){
        int p=v+half*8, q=t*16+lm;
        float val = (p<NCA)? acc[v]*inv + sig1T[(size_t)((j*NH+h)*NCA+p)*LIG+q] : 0.f;
        lg[p*32+q]=val;
      }
    }
    __syncthreads();
    float part=0.f;
    for(int idx=lane; idx<NCA*LIG; idx+=32){ int p=idx>>5,q=idx&31; part+=amr[p]*lg[p*32+q]*lmq[q]; }
    for(int s=16;s>0;s>>=1) part+=__shfl_xor(part,s,32);
    if(lane==0) r1buf[j*NH+h]=part/(sr*sc);
    // ar: softmax over q, mask, L1 renorm; record normalized-row L1 (+eps) for al
    if(lane<16){
      int p=lane;
      if(p<NCA){
        float mx=-3e38f;
        for(int q=0;q<LIG;q++) mx=fmaxf(mx,lg[p*32+q]);
        float s=0.f,s2=0.f;
        for(int q=0;q<LIG;q++){ float ex=__expf(lg[p*32+q]-mx); s+=ex; s2+=ex*lmq[q]; }
        float S = amr[p]*s2/s;
        float den0 = S + 1e-7f;
        float scale = amr[p]/(s*den0);
        for(int q=0;q<LIG;q++) arr[p*32+q]=__expf(lg[p*32+q]-mx)*lmq[q]*scale;
        rden[p] = S/den0 + 1e-7f;   // sum|ar_norm| + eps (reference uses normalized ar here)
      } else {
        for(int q=0;q<LIG;q++) arr[p*32+q]=0.f;
        rden[p]=1.f;
      }
    }
    __syncthreads();
    // al: softmax over p (col-wise), mask, divide by rden[p]; store transposed
    {
      int q=lane;
      float mx=-3e38f;
      for(int p=0;p<NCA;p++) mx=fmaxf(mx,lg[p*32+q]);
      float s=0.f;
      for(int p=0;p<NCA;p++) s+=__expf(lg[p*32+q]-mx);
      float invs=1.f/s;
      for(int p=0;p<NCA;p++) alT[q*16+p]=__expf(lg[p*32+q]-mx)*invs*amr[p]*lmq[q]/rden[p];
      alT[q*16+14]=0.f; alT[q*16+15]=0.f;
    }
    __syncthreads();
    // aVr = ar @ Vl_h  -> ssp stored
    #pragma unroll
    for(int t=0;t<2;t++){
      v8f acc=z8;
      #pragma unroll
      for(int kk=0;kk<LIG;kk+=4){
        int k0=kk+half*2;
        v2f a; a.x=arr[lm*32+k0]; a.y=arr[lm*32+k0+1];
        int n=t*16+lm;
        v2f bv; bv.x=Vlb[(size_t)(b*LIG+k0  )*HCD+h*DH+n];
                bv.y=Vlb[(size_t)(b*LIG+k0+1)*HCD+h*DH+n];
        acc=wmma4(a,bv,acc);
      }
      #pragma unroll
      for(int v=0;v<8;v++){
        int p=v+half*8;
        if(p<NCA) sspAVr[(size_t)(j*NCA+p)*HCD + h*DH + t*16+lm] = sspf(acc[v]);
      }
    }
    // aVlg[q,d] = sum_p al[p,q]*V[ri,p,h,d]  (A = al^T 32x16, B 16x32 padded)
    #pragma unroll
    for(int tmn=0; tmn<4; tmn++){
      int tmi=tmn>>1, tni=tmn&1;
      v8f acc=z8;
      #pragma unroll
      for(int kk=0;kk<16;kk+=4){
        int k0=kk+half*2;
        int mrow=tmi*16+lm;
        v2f a; a.x=alT[mrow*16+k0]; a.y=alT[mrow*16+k0+1];
        int n=tni*16+lm;
        v2f bv;
        bv.x=(k0  <NCA)? Vb[(size_t)(ri*NCA+k0  )*HCD+h*DH+n]:0.f;
        bv.y=(k0+1<NCA)? Vb[(size_t)(ri*NCA+k0+1)*HCD+h*DH+n]:0.f;
        acc=wmma4(a,bv,acc);
      }
      #pragma unroll
      for(int v=0;v<8;v++){
        int q=tmi*16+v+half*8;
        aVlg[(size_t)(j*LIG+q)*HCD + h*DH + tni*16+lm]=acc[v];
      }
    }
    __syncthreads();
  }
}

// ---------------------------------------------------------------- beta1: segment softmax of r1 over col1 (16 edges / batch)
__global__ void beta1_kernel(const float* __restrict__ r1, float* __restrict__ beta1){
  int t = threadIdx.x; if(t>=16) return;
  int b=t>>2, h=t&3;
  float mx=-3e38f;
  for(int j2=0;j2<16;j2++) mx=fmaxf(mx, r1[(b*16+j2)*NH+h]);
  float s=0.f;
  for(int j2=0;j2<16;j2++) s+=__expf(r1[(b*16+j2)*NH+h]-mx);
  float invs=1.f/s;
  for(int j2=0;j2<16;j2++) beta1[(b*16+j2)*NH+h]=__expf(r1[(b*16+j2)*NH+h]-mx)*invs;
}

// ---------------------------------------------------------------- updL (segment sum + ssp)
__global__ void updl_kernel(const float* __restrict__ aVlg, const float* __restrict__ beta1,
                            float* __restrict__ sspUL){
  int idx = blockIdx.x*blockDim.x+threadIdx.x;
  if(idx>=BSZ*LIG*HCD) return;
  int cch = idx & (HCD-1);
  int bq = idx>>7;
  int b = bq>>5, q = bq&31;
  int h = cch>>5;
  float s=0.f;
  for(int j2=0;j2<16;j2++){
    int j=b*16+j2;
    s += beta1[j*NH+h]*aVlg[(size_t)(j*LIG+q)*HCD+cch];
  }
  sspUL[idx]=sspf(s);
}

// ---------------------------------------------------------------- final atom mask over res_H3
__global__ void mask_kernel(const int* __restrict__ resS, float* __restrict__ outH){
  int row=blockIdx.x, t=threadIdx.x;
  int node=row/NCA, p=row-node*NCA;
  outH[(size_t)row*HCD+t] *= amask_of(resS[node],p);
}

// ================================================================ host
extern "C" void kernel_launch(void* const* d_in, const int* in_sizes, int n_in,
                              void* d_out, int out_size, void* d_ws, size_t ws_size,
                              hipStream_t stream)
{
  (void)in_sizes; (void)n_in; (void)out_size; (void)ws_size;
  const float* res_H   =(const float*)d_in[0];
  const float* res_X   =(const float*)d_in[1];
  const float* lig_pos =(const float*)d_in[2];
  const float* lig_feat=(const float*)d_in[3];
  const float* lig_mask=(const float*)d_in[4];
  const float* ln_g=(const float*)d_in[5];
  const float* ln_b=(const float*)d_in[6];
  const float* ln1_g=(const float*)d_in[7];
  const float* ln1_b=(const float*)d_in[8];
  const float* WQ=(const float*)d_in[9];
  const float* WK=(const float*)d_in[10];
  const float* WV=(const float*)d_in[11];
  const float* WKl=(const float*)d_in[12];
  const float* WVl=(const float*)d_in[13];
  const float* WO=(const float*)d_in[14];
  const float* WOl=(const float*)d_in[15];
  const float* WOl1=(const float*)d_in[16];
  const float* Ti_w1=(const float*)d_in[17];
  const float* Ti_b1=(const float*)d_in[18];
  const float* Ti_w2=(const float*)d_in[19];
  const float* Ti_b2=(const float*)d_in[20];
  const float* sD_w1=(const float*)d_in[21];
  const float* sD_b1=(const float*)d_in[22];
  const float* sD_w2=(const float*)d_in[23];
  const float* sD_b2=(const float*)d_in[24];
  const float* sD1_w1=(const float*)d_in[25];
  const float* sD1_b1=(const float*)d_in[26];
  const float* sD1_w2=(const float*)d_in[27];
  const float* sD1_b2=(const float*)d_in[28];
  const int*   res_S=(const int*)d_in[29];

  float* outp=(float*)d_out;
  float* outH=outp;
  float* outX=outp + (size_t)NRES*NCA*HCD;
  float* outL=outp + (size_t)NRES*NCA*HCD + (size_t)NRES*NCA*3;

  float* w=(float*)d_ws;
  auto take=[&](size_t n){ float* r=w; w+=n; return r; };
  float* resHn = take((size_t)NRES*NCA*HCD);
  float* Qb    = take((size_t)NRES*NCA*HCD);
  float* Kb    = take((size_t)NRES*NCA*HCD);
  float* Vb    = take((size_t)NRES*NCA*HCD);
  float* Tq    = take((size_t)NRES*NCA*HCD);
  float* Tk    = take((size_t)NRES*NCA*HCD);
  float* sigT  = take((size_t)NEDGE*NH*NCA*NCA);
  float* fpre  = take((size_t)NPAIR);
  float* aVb   = take((size_t)NEDGE*NCA*HCD);
  float* fXb   = take((size_t)NEDGE*NCA*3);
  float* rb    = take((size_t)NEDGE*NH);
  float* betab = take((size_t)NEDGE*NH);
  float* sspUH = take((size_t)NRES*NCA*HCD);
  float* lf    = take((size_t)BSZ*LIG*HCD);
  float* Klb   = take((size_t)BSZ*LIG*HCD);
  float* Vlb   = take((size_t)BSZ*LIG*HCD);
  float* sig1T = take((size_t)E1C*NH*NCA*LIG);
  float* sAVr  = take((size_t)E1C*NCA*HCD);
  float* aVlg  = take((size_t)E1C*LIG*HCD);
  float* r1b   = take((size_t)E1C*NH);
  float* beta1b= take((size_t)E1C*NH);
  float* sspUL = take((size_t)BSZ*LIG*HCD);
  int*   nbr   = (int*)w;

  const int M3584 = NRES*NCA;                // 3584
  const int gbig  = (M3584/16)*8/4;          // 448 blocks of 128
  const int gsml  = (BSZ*LIG/16)*8/4;        // 16
  const int gmid  = (E1C*NCA/16)*8/4;        // 112

  knn_kernel<<<1,256,0,stream>>>(res_X, nbr);
  ln_kernel<<<M3584,HCD,0,stream>>>(res_H, ln_g, ln_b, resHn);
  gemm_nt<<<gbig,128,0,stream>>>(resHn, WQ, Qb, nullptr, M3584, HCD, 0, 0);
  gemm_nt<<<gbig,128,0,stream>>>(resHn, WK, Kb, nullptr, M3584, HCD, 0, 0);
  gemm_nt<<<gbig,128,0,stream>>>(resHn, WV, Vb, nullptr, M3584, HCD, 0, 0);
  gemm_nt<<<gbig,128,0,stream>>>(Qb, Ti_w1, Tq, nullptr, M3584, 2*HCD+ECD, 0,   0);
  gemm_nt<<<gbig,128,0,stream>>>(Kb, Ti_w1, Tk, nullptr, M3584, 2*HCD+ECD, HCD, 0);
  pair_mlp<<<NPAIR/16,32,0,stream>>>(0,res_X,nullptr,nbr,sD_w1,ECD,0,sD_b1,sD_w2,sD_b2,nullptr,nullptr,sigT);
  pair_mlp<<<NPAIR/16,32,0,stream>>>(1,res_X,nullptr,nbr,Ti_w1,2*HCD+ECD,2*HCD,Ti_b1,Ti_w2,Ti_b2,Tq,Tk,fpre);
  edge_attn<<<NEDGE,32,0,stream>>>(Qb,Kb,Vb,sigT,fpre,res_X,nbr,res_S,rb,aVb,fXb);
  beta_kernel<<<1,256,0,stream>>>(rb,betab);
  updh_dx<<<NRES,HCD,0,stream>>>(aVb,betab,fXb,res_X,sspUH,outX);
  gemm_nt<<<gbig,128,0,stream>>>(sspUH, WO, outH, resHn, M3584, HCD, 0, 1);  // res_H2 (pre-mask)
  ln_kernel<<<BSZ*LIG,HCD,0,stream>>>(lig_feat, ln1_g, ln1_b, lf);
  gemm_nt<<<gsml,128,0,stream>>>(lf, WKl, Klb, nullptr, BSZ*LIG, HCD, 0, 0);
  gemm_nt<<<gsml,128,0,stream>>>(lf, WVl, Vlb, nullptr, BSZ*LIG, HCD, 0, 0);
  pair_mlp<<<NP1/16,32,0,stream>>>(2,outX,lig_pos,nullptr,sD1_w1,ECD,0,sD1_b1,sD1_w2,sD1_b2,nullptr,nullptr,sig1T);
  lig_attn<<<E1C,32,0,stream>>>(Qb,Klb,Vlb,Vb,sig1T,lig_mask,res_S,r1b,sAVr,aVlg);
  beta1_kernel<<<1,32,0,stream>>>(r1b,beta1b);
  gemm_nt<<<gmid,128,0,stream>>>(sAVr, WOl, outH, nullptr, E1C*NCA, HCD, 0, 2); // scatter-add
  updl_kernel<<<BSZ*LIG*HCD/128,128,0,stream>>>(aVlg,beta1b,sspUL);
  gemm_nt<<<gsml,128,0,stream>>>(sspUL, WOl1, outL, lf, BSZ*LIG, HCD, 0, 1);   // lf2
  mask_kernel<<<M3584,HCD,0,stream>>>(res_S,outH);
}